// OffsetAlignment_67980742361806
// MI455X (gfx1250) — compile-verified
//
#include <hip/hip_runtime.h>
#include <math.h>

#define B_ 4
#define C_ 64
#define H_ 256
#define W_ 448
#define HW_ (H_*W_)
#define G_ 8
#define CPG_ 8
#define NPG_ (CPG_*HW_)          // 917504 elements per (b,group), contiguous in NCHW
#define BLK1_PER_G 112           // blocks per group for stats
#define ELEM_PER_BLK1 8192       // 112*8192 = 917504

typedef __attribute__((ext_vector_type(2))) float v2f;
typedef __attribute__((ext_vector_type(8))) float v8f;

// ---------------- Kernel 1: per-(b,group) partial sums (deterministic) ----------------
__launch_bounds__(256)
__global__ void gn_stats_kernel(const float* __restrict__ x,
                                double* __restrict__ psum, double* __restrict__ psq) {
    const int bg = blockIdx.y;                       // 0..31  (b*8+g)
    const float* base = x + (size_t)bg * NPG_;       // group channels are contiguous
    const int start = blockIdx.x * ELEM_PER_BLK1;
    float s = 0.f, s2 = 0.f;
#pragma unroll 4
    for (int i = 0; i < 32; ++i) {
        float v = base[start + i * 256 + threadIdx.x];
        s += v; s2 += v * v;
    }
    __shared__ double sd[256], sq[256];
    sd[threadIdx.x] = (double)s; sq[threadIdx.x] = (double)s2;
    __syncthreads();
    for (int off = 128; off > 0; off >>= 1) {
        if (threadIdx.x < off) { sd[threadIdx.x] += sd[threadIdx.x + off];
                                 sq[threadIdx.x] += sq[threadIdx.x + off]; }
        __syncthreads();
    }
    if (threadIdx.x == 0) {
        psum[bg * BLK1_PER_G + blockIdx.x] = sd[0];
        psq [bg * BLK1_PER_G + blockIdx.x] = sq[0];
    }
}

// ---------------- Kernel 2: finalize -> per-(b,c) scale/shift ----------------
__launch_bounds__(128)
__global__ void gn_finalize_kernel(const double* __restrict__ psum, const double* __restrict__ psq,
                                   const float* __restrict__ gnw, const float* __restrict__ gnb,
                                   float* __restrict__ scale, float* __restrict__ shift) {
    const int bg = blockIdx.x, b = bg >> 3, g = bg & 7;
    __shared__ double sd[128], sq[128];
    const int t = threadIdx.x;
    sd[t] = (t < BLK1_PER_G) ? psum[bg * BLK1_PER_G + t] : 0.0;
    sq[t] = (t < BLK1_PER_G) ? psq [bg * BLK1_PER_G + t] : 0.0;
    __syncthreads();
    for (int off = 64; off > 0; off >>= 1) {
        if (t < off) { sd[t] += sd[t + off]; sq[t] += sq[t + off]; }
        __syncthreads();
    }
    __shared__ float smean, srstd;
    if (t == 0) {
        double mean = sd[0] / (double)NPG_;
        double var  = sq[0] / (double)NPG_ - mean * mean;
        smean = (float)mean;
        srstd = (float)(1.0 / sqrt(var + 1e-5));
    }
    __syncthreads();
    if (t < CPG_) {
        int c = g * CPG_ + t;
        float sc = srstd * gnw[c];
        scale[b * C_ + c] = sc;
        shift[b * C_ + c] = gnb[c] - smean * sc;
    }
}

// ---------------- Kernel 3: fused on-the-fly groupnorm + 8-neighbor cosine sim ----------------
__launch_bounds__(256)
__global__ void sim_kernel(const float* __restrict__ x,
                           const float* __restrict__ scale, const float* __restrict__ shift,
                           float* __restrict__ sim) {
    const int tx = threadIdx.x & 15, ty = threadIdx.x >> 4;
    const int x0 = blockIdx.x * 16, y0 = blockIdx.y * 16, b = blockIdx.z;
    __shared__ float s_x[32][20][20];        // [c][y][x], halo 2 (dil=2), conflict-free
    __shared__ float s_sc[C_], s_sh[C_];
    if (threadIdx.x < C_)            s_sc[threadIdx.x]       = scale[b * C_ + threadIdx.x];
    else if (threadIdx.x < 2 * C_)   s_sh[threadIdx.x - C_]  = shift[b * C_ + threadIdx.x - C_];

    const int dyv[8] = {-2,-2,-2, 0, 0, 2, 2, 2};
    const int dxv[8] = {-2, 0, 2,-2, 2,-2, 0, 2};
    float censq = 0.f;
    float dot[8] = {0,0,0,0,0,0,0,0};
    float nsq[8] = {0,0,0,0,0,0,0,0};

    for (int c0 = 0; c0 < C_; c0 += 32) {
        __syncthreads();                      // also covers s_sc/s_sh on first pass
        for (int idx = threadIdx.x; idx < 32 * 400; idx += 256) {
            int c = idx / 400, rem = idx % 400, yy = rem / 20, xx = rem % 20;
            int gy = y0 + yy - 2, gx = x0 + xx - 2;
            float v = 0.f;                    // zero padding = raw zeros (pad is post-norm)
            if (gy >= 0 && gy < H_ && gx >= 0 && gx < W_) {
                int cc = c0 + c;
                v = x[(((size_t)b * C_ + cc) * H_ + gy) * W_ + gx] * s_sc[cc] + s_sh[cc];
            }
            s_x[c][yy][xx] = v;
        }
        __syncthreads();
        for (int c = 0; c < 32; ++c) {
            float cen = s_x[c][ty + 2][tx + 2];
            censq += cen * cen;
#pragma unroll
            for (int n = 0; n < 8; ++n) {
                float nb = s_x[c][ty + 2 + dyv[n]][tx + 2 + dxv[n]];
                dot[n] += cen * nb;
                nsq[n] += nb * nb;
            }
        }
    }
    float nc = fmaxf(sqrtf(censq), 1e-8f);
    int yq = y0 + ty, xq = x0 + tx;
#pragma unroll
    for (int n = 0; n < 8; ++n) {
        float s = dot[n] / (nc * fmaxf(sqrtf(nsq[n]), 1e-8f));
        sim[(((size_t)b * 8 + n) * H_ + yq) * W_ + xq] = s;
    }
}

// ---------------- Kernel 4: both 3x3 convs as one implicit GEMM on WMMA f32 16x16x4 ----------------
// M = 16 pixels/wave, K = 72 (8ch x 9taps) in 18 chunks of 4, N cols 0..3 = {offx,offy,scalex,scaley}.
// A-panel im2col'd into LDS so the WMMA loop is pure immediate-offset ds_loads + wmma.
#define APAD 72                  // A row stride (floats): 2*72 % 64 == 16 -> half-wave bank split
#define BPAD 74                  // B row stride (floats): 8B-aligned pairs, 10n%64 distinct n<16
__launch_bounds__(128)
__global__ void conv_offset_kernel(const float* __restrict__ sim,
                                   const float* __restrict__ ow, const float* __restrict__ ob,
                                   const float* __restrict__ sw, const float* __restrict__ sb,
                                   float* __restrict__ offs) {
    const int tid = threadIdx.x, lane = tid & 31, wave = tid >> 5;
    const int b = blockIdx.z, y = blockIdx.y, xbase = blockIdx.x * 64;
    __shared__ float s_sim[8][3][66];               // 8 ch, rows y-1..y+1, 64 px + halo 1
    __shared__ __align__(16) float s_A[72][APAD];   // im2col panel: [k][pixel]
    __shared__ __align__(16) float s_wt[16][BPAD];  // B matrix, zero-padded to 16 cols
    __shared__ float s_out[4][4][16];               // [wave][col][pixel]

    // B table: s_wt[n][k] = weight(out n, k=ch*9+tap); rows 4..15 and k>=72 are zero.
    for (int i = tid; i < 16 * BPAD; i += 128) {
        int n = i / BPAD, k = i % BPAD;
        float v = 0.f;
        if (k < 72) {
            if (n < 2)      v = ow[n * 72 + k];
            else if (n < 4) v = sw[(n - 2) * 72 + k];
        }
        (&s_wt[0][0])[i] = v;
    }
    // sim tile with halo (zero-padded SAME boundary)
    for (int i = tid; i < 8 * 3 * 66; i += 128) {
        int ch = i / 198, rem = i % 198, r = rem / 66, xx = rem % 66;
        int gy = y + r - 1, gx = xbase + xx - 1;
        float v = 0.f;
        if (gy >= 0 && gy < H_ && gx >= 0 && gx < W_)
            v = sim[(((size_t)b * 8 + ch) * H_ + gy) * W_ + gx];
        s_sim[ch][r][xx] = v;
    }
    __syncthreads();
    // Expand to im2col A-panel: s_A[k][px] = s_sim[k/9][(k%9)/3][px + (k%9)%3]
    for (int i = tid; i < 72 * 64; i += 128) {
        int k = i >> 6, px = i & 63;
        int ch = k / 9, t = k % 9;
        s_A[k][px] = s_sim[ch][t / 3][px + t % 3];
    }
    __syncthreads();

    // ISA A layout (16x4 f32): lanes 0-15 -> M=lane, {K0,K1}; lanes 16-31 -> M=lane-16, {K2,K3}.
    // B layout mirrored on K: lanes 0-15 rows {0,1}, lanes 16-31 rows {2,3}, col = lane&15.
    const int m   = lane & 15;
    const int kof = (lane < 16) ? 0 : 2;
    const int px  = wave * 16 + m;
    const int n   = lane & 15;
    const float* pA = &s_A[0][0] + (size_t)kof * APAD + px;   // +j*4*APAD, +APAD per K step
    const float* pB = &s_wt[0][0] + (size_t)n * BPAD + kof;   // +j*4 (8B-aligned pair)
    v8f acc = {};
#pragma unroll
    for (int j = 0; j < 18; ++j) {
        v2f a, bb;
        a.x = pA[j * 4 * APAD];
        a.y = pA[j * 4 * APAD + APAD];
        bb  = *(const v2f*)(pB + j * 4);
        acc = __builtin_amdgcn_wmma_f32_16x16x4_f32(false, a, false, bb,
                                                    (short)0, acc, false, false);
    }
    // D layout: VGPR r holds (M=r, N=lane) for lanes 0-15 and (M=r+8, N=lane-16) for 16-31.
    if ((lane & 15) < 4) {
        int col = lane & 15, rbase = (lane < 16) ? 0 : 8;
#pragma unroll
        for (int r = 0; r < 8; ++r) s_out[wave][col][rbase + r] = acc[r];
    }
    __syncthreads();
    if (tid < 64) {
        int w = tid >> 4, mm = tid & 15;
        float o0 = s_out[w][0][mm] + ob[0];
        float o1 = s_out[w][1][mm] + ob[1];
        float s0 = s_out[w][2][mm] + sb[0];
        float s1 = s_out[w][3][mm] + sb[1];
        float offx = o0 * (1.f / (1.f + __expf(-s0)));
        float offy = o1 * (1.f / (1.f + __expf(-s1)));
        int gx = xbase + w * 16 + mm;
        offs[(((size_t)b * 2 + 0) * H_ + y) * W_ + gx] = offx;
        offs[(((size_t)b * 2 + 1) * H_ + y) * W_ + gx] = offy;
    }
}

// ---------------- Kernel 5: nearest resample with on-the-fly groupnorm ----------------
__launch_bounds__(256)
__global__ void sample_kernel(const float* __restrict__ x, const float* __restrict__ offs,
                              const float* __restrict__ scale, const float* __restrict__ shift,
                              float* __restrict__ out) {
    const int b = blockIdx.y;
    const int p = blockIdx.x * 256 + threadIdx.x;   // HW divisible by 256
    const int y = p / W_, xq = p % W_;
    __shared__ float s_sc[C_], s_sh[C_];
    if (threadIdx.x < C_)          s_sc[threadIdx.x]      = scale[b * C_ + threadIdx.x];
    else if (threadIdx.x < 2 * C_) s_sh[threadIdx.x - C_] = shift[b * C_ + threadIdx.x - C_];
    __syncthreads();
    float offx = offs[(((size_t)b * 2 + 0) * H_ + y) * W_ + xq];
    float offy = offs[(((size_t)b * 2 + 1) * H_ + y) * W_ + xq];
    float gx  = (xq + 0.5f) + offx;
    float gy  = (y  + 0.5f) + offy;
    float ngx = 2.0f * gx / (float)W_ - 1.0f;
    float ngy = 2.0f * gy / (float)H_ - 1.0f;
    float fx  = ((ngx + 1.0f) * (float)W_ - 1.0f) * 0.5f;
    float fy  = ((ngy + 1.0f) * (float)H_ - 1.0f) * 0.5f;
    int ix = min(max((int)rintf(fx), 0), W_ - 1);   // rintf = round-half-even (jnp.round)
    int iy = min(max((int)rintf(fy), 0), H_ - 1);
    const float* src = x   + (size_t)b * C_ * HW_ + (size_t)iy * W_ + ix;
    float*       dst = out + (size_t)b * C_ * HW_ + (size_t)y  * W_ + xq;
#pragma unroll 4
    for (int c = 0; c < C_; ++c)
        dst[(size_t)c * HW_] = src[(size_t)c * HW_] * s_sc[c] + s_sh[c];
}

extern "C" void kernel_launch(void* const* d_in, const int* in_sizes, int n_in,
                              void* d_out, int out_size, void* d_ws, size_t ws_size,
                              hipStream_t stream) {
    const float* aligned = (const float*)d_in[0];
    // d_in[1] original_feat, d_in[2] flow: unused by the reference
    const float* gnw = (const float*)d_in[3];
    const float* gnb = (const float*)d_in[4];
    const float* ow  = (const float*)d_in[5];
    const float* ob  = (const float*)d_in[6];
    const float* sw  = (const float*)d_in[7];
    const float* sb  = (const float*)d_in[8];
    float* out = (float*)d_out;

    char* ws = (char*)d_ws;
    double* psum  = (double*)ws;                       // 3584 doubles
    double* psq   = psum + 32 * BLK1_PER_G;            // 3584 doubles (-> 57344 B)
    float*  scale = (float*)(ws + 57344);              // 256 floats
    float*  shift = scale + 256;
    float*  sim   = (float*)(ws + 65536);              // B*8*H*W floats (14.68 MB)
    float*  offs  = sim + (size_t)B_ * 8 * HW_;        // B*2*H*W floats (3.67 MB)

    gn_stats_kernel   <<<dim3(BLK1_PER_G, 32), 256, 0, stream>>>(aligned, psum, psq);
    gn_finalize_kernel<<<32, 128, 0, stream>>>(psum, psq, gnw, gnb, scale, shift);
    sim_kernel        <<<dim3(W_/16, H_/16, B_), 256, 0, stream>>>(aligned, scale, shift, sim);
    conv_offset_kernel<<<dim3(W_/64, H_, B_), 128, 0, stream>>>(sim, ow, ob, sw, sb, offs);
    sample_kernel     <<<dim3(HW_/256, B_), 256, 0, stream>>>(aligned, offs, scale, shift, out);
}